// CompressedAttention_81372450390152
// MI455X (gfx1250) — compile-verified
//
#include <hip/hip_runtime.h>
#include <math.h>

// CDNA5 / gfx1250: wave32, WMMA 16x16x4 f32
typedef __attribute__((ext_vector_type(2))) float v2f;
typedef __attribute__((ext_vector_type(8))) float v8f;

#define BATCH 4
#define SEQ   4096
#define DM    5
#define NKT   (SEQ / 16)      // 256 key tiles of 16
#define QTPB  8               // query tiles (waves) per block
#define EPSV  1e-6f
#define RMS_S 16.0f

// ds_swizzle group-of-32 XOR butterfly: offset = (xor_mask<<10) | and_mask(0x1F)
template <int IMM>
__device__ __forceinline__ float swz(float v) {
    return __int_as_float(__builtin_amdgcn_ds_swizzle(__float_as_int(v), IMM));
}
// xor masks 1,2,4,8 stay within each 16-lane half (row group)
#define BFLY_MAX(x)                         \
    x = fmaxf(x, swz<0x041f>(x));           \
    x = fmaxf(x, swz<0x081f>(x));           \
    x = fmaxf(x, swz<0x101f>(x));           \
    x = fmaxf(x, swz<0x201f>(x));
#define BFLY_SUM(x)                         \
    x += swz<0x041f>(x);                    \
    x += swz<0x081f>(x);                    \
    x += swz<0x101f>(x);                    \
    x += swz<0x201f>(x);

__global__ __launch_bounds__(256) void
CompressedAttention_gfx1250_kernel(const float* __restrict__ x,
                                   const float* __restrict__ fcos,
                                   const float* __restrict__ fsin,
                                   const float* __restrict__ q_u,
                                   const float* __restrict__ k_u,
                                   float* __restrict__ out)
{
    // {k_rot0, k_rot1, v, 0} per key position of this batch: 64 KB LDS
    __shared__ float4 kv[SEQ];

    const int b    = blockIdx.x >> 5;      // 32 blocks per batch element
    const int grp  = blockIdx.x & 31;      // which group of 8 query tiles
    const int tid  = threadIdx.x;
    const int wave = tid >> 5;
    const int lane = tid & 31;
    const int col  = lane & 15;            // score-tile column owned by this lane
    const int hi   = lane >> 4;            // 0: rows 0-7, 1: rows 8-15 of D

    // ---------------- stage K/V for batch b into LDS (recompute: cheap) ----
    const float ku0 = k_u[0], ku1 = k_u[1];
    for (int i = tid; i < SEQ; i += 256) {
        float x0 = x[(b * SEQ + i) * DM + 0];
        float x1 = x[(b * SEQ + i) * DM + 1];
        float c  = fcos[i], s = fsin[i];
        float k0 = x0 * ku0, k1 = x0 * ku1;
        float inv = RMS_S * rsqrtf(0.5f * (k0 * k0 + k1 * k1) + EPSV);
        k0 *= inv; k1 *= inv;
        kv[i] = make_float4(k0 * c - k1 * s, k0 * s + k1 * c, x1, 0.0f);
    }
    __syncthreads();

    // ---------------- per-wave: 16 queries, both heads ----------------------
    const int s0 = (grp * QTPB + wave) * 16;

    // A matrices (16x4, K=2..3 zero): lane<16 holds row M=lane, {K0,K1}.
    // Scale folds 1/sqrt(D) AND log2(e): scores land in log2 domain so the
    // softmax can use raw v_exp_f32 (exp2) with no per-score multiply.
    v2f a0 = {0.0f, 0.0f}, a1 = {0.0f, 0.0f};
    if (lane < 16) {
        const int sq = s0 + lane;
        const float x0 = x[(b * SEQ + sq) * DM + 0];
        const float c = fcos[sq], sn = fsin[sq];
        const float sc = 0.70710678119f * 1.44269504089f; // log2e / sqrt(2)
        float q0 = x0 * q_u[0], q1 = x0 * q_u[1];
        float inv = RMS_S * rsqrtf(0.5f * (q0 * q0 + q1 * q1) + EPSV);
        q0 *= inv; q1 *= inv;
        a0[0] = (q0 * c - q1 * sn) * sc;
        a0[1] = (q0 * sn + q1 * c) * sc;
        q0 = x0 * q_u[2]; q1 = x0 * q_u[3];
        inv = RMS_S * rsqrtf(0.5f * (q0 * q0 + q1 * q1) + EPSV);
        q0 *= inv; q1 *= inv;
        a1[0] = (q0 * c - q1 * sn) * sc;
        a1[1] = (q0 * sn + q1 * c) * sc;
    }

    // ---------------- pass 1: row maxima (no exp) ---------------------------
    float m0[8], m1[8];
#pragma unroll
    for (int r = 0; r < 8; ++r) { m0[r] = -3.0e38f; m1[r] = -3.0e38f; }

    for (int kt = 0; kt < NKT; ++kt) {
        float4 k4 = kv[kt * 16 + col];
        v2f bb;                                   // B rows K=0,1 live in lo half
        bb[0] = hi ? 0.0f : k4.x;
        bb[1] = hi ? 0.0f : k4.y;
        v8f cz = {};
        v8f t0 = __builtin_amdgcn_wmma_f32_16x16x4_f32(
            false, a0, false, bb, (short)0, cz, false, false);
        v8f t1 = __builtin_amdgcn_wmma_f32_16x16x4_f32(
            false, a1, false, bb, (short)0, cz, false, false);
#pragma unroll
        for (int r = 0; r < 8; ++r) {
            m0[r] = fmaxf(m0[r], t0[r]);
            m1[r] = fmaxf(m1[r], t1[r]);
        }
    }
    // butterfly max across the 16 lanes sharing each row
#pragma unroll
    for (int r = 0; r < 8; ++r) { BFLY_MAX(m0[r]); BFLY_MAX(m1[r]); }

    // ---------------- pass 2: exp2 + accumulate (1 v_exp per score) ---------
    float l0[8], l1[8], ac0[8], ac1[8];
#pragma unroll
    for (int r = 0; r < 8; ++r) { l0[r] = l1[r] = ac0[r] = ac1[r] = 0.0f; }

    for (int kt = 0; kt < NKT; ++kt) {
        float4 k4 = kv[kt * 16 + col];
        v2f bb;
        bb[0] = hi ? 0.0f : k4.x;
        bb[1] = hi ? 0.0f : k4.y;
        const float vv = k4.z;                    // v for key (kt*16 + col)
        v8f cz = {};
        v8f t0 = __builtin_amdgcn_wmma_f32_16x16x4_f32(
            false, a0, false, bb, (short)0, cz, false, false);
        v8f t1 = __builtin_amdgcn_wmma_f32_16x16x4_f32(
            false, a1, false, bb, (short)0, cz, false, false);
#pragma unroll
        for (int r = 0; r < 8; ++r) {
            float p0 = __builtin_amdgcn_exp2f(t0[r] - m0[r]);  // v_exp_f32
            float p1 = __builtin_amdgcn_exp2f(t1[r] - m1[r]);
            l0[r]  += p0;           l1[r]  += p1;
            ac0[r] += p0 * vv;      ac1[r] += p1 * vv;
        }
    }
    // butterfly sums across the 16 lanes of each row
#pragma unroll
    for (int r = 0; r < 8; ++r) {
        BFLY_SUM(l0[r]);  BFLY_SUM(ac0[r]);
        BFLY_SUM(l1[r]);  BFLY_SUM(ac1[r]);
    }

    // ---------------- write: channel 2 = head0 + head1; others 0 -----------
#pragma unroll
    for (int r = 0; r < 8; ++r) {
        if (col == r) {
            const int sq = s0 + r + 8 * hi;
            float res = ac0[r] / l0[r] + ac1[r] / l1[r];
            float* o = out + (b * SEQ + sq) * DM;
            o[0] = 0.0f; o[1] = 0.0f; o[2] = res; o[3] = 0.0f; o[4] = 0.0f;
        }
    }
}

extern "C" void kernel_launch(void* const* d_in, const int* in_sizes, int n_in,
                              void* d_out, int out_size, void* d_ws, size_t ws_size,
                              hipStream_t stream) {
    (void)in_sizes; (void)n_in; (void)out_size; (void)d_ws; (void)ws_size;
    const float* x    = (const float*)d_in[0];   // [4,4096,5]
    const float* fcos = (const float*)d_in[1];   // [4096,1]
    const float* fsin = (const float*)d_in[2];   // [4096,1]
    const float* q_u  = (const float*)d_in[3];   // [4]
    const float* k_u  = (const float*)d_in[4];   // [2]
    float* out = (float*)d_out;                  // [4,4096,5]

    // 4 batches * 32 groups = 128 blocks; 8 waves/block, 1 query-tile+2 heads/wave
    dim3 grid(BATCH * 32), block(256);
    CompressedAttention_gfx1250_kernel<<<grid, block, 0, stream>>>(
        x, fcos, fsin, q_u, k_u, out);
}